// SoftRasterizer_82712480186437
// MI455X (gfx1250) — compile-verified
//
#include <hip/hip_runtime.h>

typedef __attribute__((ext_vector_type(2))) float v2f;
typedef __attribute__((ext_vector_type(8))) float v8f;

#define IMG        128
#define NFACES     1024
#define NEAR_P     1.0f
#define FAR_P      100.0f
#define EPS_P      0.001f
#define INV_SIGMA  1.0e5f          // 1/1e-5
#define INV_GAMMA  1.0e4f          // 1/1e-4
#define ZPN_SCALE  (1.0f/99.0f)    // 1/(FAR-NEAR)
#define THRESH     9.210240371976183e-05f  // SIGMA*ln(1/DIST_EPS - 1)

#define FPC   128     // faces per LDS chunk
#define REC   22      // floats per face record in LDS
#define NCHUNK (NFACES / FPC)

#define RCP(x) __builtin_amdgcn_rcpf(x)   // single v_rcp_f32, ~1 ulp

// face record: [0..5] x0 y0 x1 y1 x2 y2 | [6] invdet | [7..9] 1/z0..1/z2
//              [10..12] invL(01,12,20)  | [13..21] tex[3][3]

__device__ __forceinline__ float seg_d2(float px, float py, float ax, float ay,
                                        float bx, float by, float invL) {
  float ex = bx - ax, ey = by - ay;
  float dx = px - ax, dy = py - ay;
  float t  = fminf(fmaxf((dx * ex + dy * ey) * invL, 0.f), 1.f);
  float qx = dx - t * ex, qy = dy - t * ey;
  return qx * qx + qy * qy;
}

template <bool PASS2>
__device__ __forceinline__ void eval_face(const float* __restrict__ r,
                                          float px, float py, float m,
                                          float& zpn_out, float& wf,
                                          float& c0o, float& c1o, float& c2o,
                                          float& one_minus_dmask) {
  float x0 = r[0], y0 = r[1], x1 = r[2], y1 = r[3], x2 = r[4], y2 = r[5];
  float invdet = r[6], iz0 = r[7], iz1 = r[8], iz2 = r[9];
  float e1x = x1 - x0, e1y = y1 - y0, e2x = x2 - x0, e2y = y2 - y0;
  float dx = px - x0, dy = py - y0;
  float w1 = (dx * e2y - dy * e2x) * invdet;
  float w2 = (e1x * dy - e1y * dx) * invdet;
  float w0 = 1.f - w1 - w2;
  bool inside = (w0 >= 0.f) && (w1 >= 0.f) && (w2 >= 0.f);
  float dis = fminf(seg_d2(px, py, x0, y0, x1, y1, r[10]),
              fminf(seg_d2(px, py, x1, y1, x2, y2, r[11]),
                    seg_d2(px, py, x2, y2, x0, y0, r[12])));
  bool near_ok = inside || (dis <= THRESH);
  float c0 = fminf(fmaxf(w0, 0.f), 1.f);
  float c1 = fminf(fmaxf(w1, 0.f), 1.f);
  float c2 = fminf(fmaxf(w2, 0.f), 1.f);
  float s    = fmaxf(c0 + c1 + c2, 1e-5f);
  float invs = RCP(s);
  c0 *= invs; c1 *= invs; c2 *= invs;
  float invz = c0 * iz0 + c1 * iz1 + c2 * iz2;
  float zp   = RCP(fmaxf(invz, 1e-8f));
  bool zok   = (zp >= NEAR_P) && (zp <= FAR_P);
  bool valid = near_ok && zok;
  float zpn  = (FAR_P - zp) * ZPN_SCALE;
  zpn_out = valid ? zpn : -1.f;
  if (PASS2) {
    float x  = (inside ? 1.f : -1.f) * dis * INV_SIGMA;
    float e  = __expf(-fabsf(x));          // stable sigmoid
    float sp = RCP(1.f + e);
    float dfrag = (x >= 0.f) ? sp : e * sp;
    float arg = fminf((zpn - m) * INV_GAMMA, 0.f);
    wf  = valid ? dfrag * __expf(arg) : 0.f;
    c0o = c0; c1o = c1; c2o = c2;
    one_minus_dmask = near_ok ? (1.f - dfrag) : 1.f;  // FILL_BACK=True
  }
}

__device__ __forceinline__ void load_chunk(float* __restrict__ sf,
                                           const float* __restrict__ fv,
                                           const float* __restrict__ tex,
                                           int b, int chunk, int tid) {
  if (tid < FPC) {
    int fi = chunk * FPC + tid;
    const float* fvp = fv  + ((size_t)b * NFACES + fi) * 9;
    const float* tp  = tex + ((size_t)b * NFACES + fi) * 9;
    float x0 = fvp[0], y0 = fvp[1], z0 = fvp[2];
    float x1 = fvp[3], y1 = fvp[4], z1 = fvp[5];
    float x2 = fvp[6], y2 = fvp[7], z2 = fvp[8];
    float e1x = x1 - x0, e1y = y1 - y0, e2x = x2 - x0, e2y = y2 - y0;
    float det = e1x * e2y - e1y * e2x;
    det = (fabsf(det) < 1e-10f) ? 1e-10f : det;
    float* r = &sf[tid * REC];
    r[0] = x0; r[1] = y0; r[2] = x1; r[3] = y1; r[4] = x2; r[5] = y2;
    r[6] = RCP(det);
    r[7] = RCP(z0); r[8] = RCP(z1); r[9] = RCP(z2);
    r[10] = RCP(fmaxf(e1x * e1x + e1y * e1y, 1e-12f));
    float ex = x2 - x1, ey = y2 - y1;
    r[11] = RCP(fmaxf(ex * ex + ey * ey, 1e-12f));
    ex = x0 - x2; ey = y0 - y2;
    r[12] = RCP(fmaxf(ex * ex + ey * ey, 1e-12f));
#pragma unroll
    for (int j = 0; j < 9; ++j) r[13 + j] = tp[j];
  }
}

__global__ __launch_bounds__(256)
void softras_fwd(const float* __restrict__ fv, const float* __restrict__ tex,
                 float* __restrict__ out) {
  __shared__ float s_face[FPC * REC];
  __shared__ float s_stage[8][256];

  const int tid   = threadIdx.x;
  const int lane  = tid & 31;
  const int wave  = tid >> 5;
  const int col   = lane & 15;   // pixel-in-wave (A rows / C cols per layout)
  const int klane = lane >> 4;   // which half of the K slots this lane owns

  const int gpix = blockIdx.x * 128 + wave * 16 + col;   // 128 pixels / block
  const int b    = gpix >> 14;                           // /(128*128)
  const int rem  = gpix & 16383;
  const int h    = rem >> 7;
  const int w    = rem & 127;
  const float px = (2.f * (float)w + 1.f - (float)IMG) * (1.f / (float)IMG);
  const float py = -((2.f * (float)h + 1.f - (float)IMG) * (1.f / (float)IMG));

  // hoisted B-column constants for this lane (branchless construction)
  const bool  is_rgb  = (col < 3);
  const int   cc      = is_rgb ? col : 0;        // clamped, always in-bounds
  const float onesel  = (col == 3) ? 1.f : 0.f;  // ones column -> sum(wfull)

  // ---------------- pass 1: per-pixel max of valid zp_norm ----------------
  float mloc = -1.f;
  for (int ch = 0; ch < NCHUNK; ++ch) {
    __syncthreads();
    load_chunk(s_face, fv, tex, b, ch, tid);
    __syncthreads();
#pragma unroll 2
    for (int g = 0; g < FPC / 4; ++g) {
      int f0 = g * 4 + klane * 2;
      float z0v, z1v, d0, d1, d2, d3;
      eval_face<false>(&s_face[(f0 + 0) * REC], px, py, 0.f, z0v, d0, d1, d2, d3, d0);
      mloc = fmaxf(mloc, z0v);
      eval_face<false>(&s_face[(f0 + 1) * REC], px, py, 0.f, z1v, d0, d1, d2, d3, d0);
      mloc = fmaxf(mloc, z1v);
    }
  }
  float mo = __shfl_xor(mloc, 16, 32);
  const float m = fmaxf(fmaxf(mloc, mo), EPS_P);

  // ---------------- pass 2: WMMA color/weight accumulation ----------------
  v8f acc = {};        // C: 16 pixels x 16 cols (cols 0..2 rgb, col 3 = sum w)
  float aprod = 1.f;   // prod(1 - Dmask) over this lane's faces
  for (int ch = 0; ch < NCHUNK; ++ch) {
    __syncthreads();
    load_chunk(s_face, fv, tex, b, ch, tid);
    __syncthreads();
#pragma unroll 2
    for (int g = 0; g < FPC / 4; ++g) {
      int f0 = g * 4 + klane * 2;           // this lane's two K-slots
      const float* r0 = &s_face[(f0 + 0) * REC];
      const float* r1 = &s_face[(f0 + 1) * REC];

      // unconditional (in-bounds) LDS loads for B, selected later via cndmask
      float t00 = r0[13 + 0 + cc], t01 = r1[13 + 0 + cc];
      float t10 = r0[13 + 3 + cc], t11 = r1[13 + 3 + cc];
      float t20 = r0[13 + 6 + cc], t21 = r1[13 + 6 + cc];

      float zdum, wfa, c0a, c1a, c2a, oma;
      float wfb, c0b, c1b, c2b, omb;
      eval_face<true>(r0, px, py, m, zdum, wfa, c0a, c1a, c2a, oma);
      eval_face<true>(r1, px, py, m, zdum, wfb, c0b, c1b, c2b, omb);
      aprod *= oma * omb;

      v2f a0 = {wfa * c0a, wfb * c0b};
      v2f a1 = {wfa * c1a, wfb * c1b};
      v2f a2 = {wfa * c2a, wfb * c2b};

      v2f b0, b1, b2;
      b0.x = is_rgb ? t00 : onesel;  b0.y = is_rgb ? t01 : onesel;
      b1.x = is_rgb ? t10 : onesel;  b1.y = is_rgb ? t11 : onesel;
      b2.x = is_rgb ? t20 : onesel;  b2.y = is_rgb ? t21 : onesel;

      acc = __builtin_amdgcn_wmma_f32_16x16x4_f32(false, a0, false, b0,
                                                  (short)0, acc, false, false);
      acc = __builtin_amdgcn_wmma_f32_16x16x4_f32(false, a1, false, b1,
                                                  (short)0, acc, false, false);
      acc = __builtin_amdgcn_wmma_f32_16x16x4_f32(false, a2, false, b2,
                                                  (short)0, acc, false, false);
    }
  }
  aprod *= __shfl_xor(aprod, 16, 32);

  // stage C through LDS: element (M = i + 8*klane, N = col) in vgpr i
#pragma unroll
  for (int i = 0; i < 8; ++i)
    s_stage[wave][(i + 8 * klane) * 16 + col] = acc[i];
  __syncthreads();

  if (lane < 16) {
    const float* st = &s_stage[wave][lane * 16];
    float cr = st[0], cg = st[1], cb = st[2], wsum = st[3];
    float wbg = __expf((EPS_P - m) * INV_GAMMA);
    float inv = RCP(wsum + wbg);              // BG color is (0,0,0)
    size_t base = (size_t)b * 4 * (IMG * IMG);
    int hw = h * IMG + w;
    out[base + 0 * (IMG * IMG) + hw] = cr * inv;
    out[base + 1 * (IMG * IMG) + hw] = cg * inv;
    out[base + 2 * (IMG * IMG) + hw] = cb * inv;
    out[base + 3 * (IMG * IMG) + hw] = 1.f - aprod;
  }
}

extern "C" void kernel_launch(void* const* d_in, const int* in_sizes, int n_in,
                              void* d_out, int out_size, void* d_ws, size_t ws_size,
                              hipStream_t stream) {
  (void)in_sizes; (void)n_in; (void)d_ws; (void)ws_size; (void)out_size;
  const float* fv  = (const float*)d_in[0];  // [2,1024,3,3]
  const float* tex = (const float*)d_in[1];  // [2,1024,3,3]
  float* out = (float*)d_out;                // [2,4,128,128]
  const int total_pixels = 2 * IMG * IMG;    // 32768
  dim3 grid(total_pixels / 128), block(256);
  softras_fwd<<<grid, block, 0, stream>>>(fv, tex, out);
}